// multi_attention_10007273799952
// MI455X (gfx1250) — compile-verified
//
#include <hip/hip_runtime.h>
#include <hip/hip_bf16.h>

// Problem constants (fixed by the reference)
#define GDIM 2048
#define BDIM 8
#define HDIM 8
#define LOG2E 1.4426950408889634f

typedef float v2f __attribute__((ext_vector_type(2)));
typedef float v8f __attribute__((ext_vector_type(8)));

__global__ void zero_out_kernel(float* __restrict__ out, int n) {
    int i = blockIdx.x * blockDim.x + threadIdx.x;
    if (i < n) out[i] = 0.0f;
}

// One block per (i-chunk, b*h). 256 threads = 8 wave32s.
// Each wave owns two 16-row i-tiles and loops over all 128 j-tiles.
// Score tile S^T[j,i] = k2_j * q_i produced by V_WMMA_F32_16X16X4_F32
// (rank-1: only K=0 lane of A/B populated). C-layout: lane = i (mod 16),
// VGPR r = j-row (r, or r+8 for lanes 16..31).
__global__ __launch_bounds__(256) void attn_rank1_softmax_kernel(
    const float* __restrict__ x,   // (B, G)
    const float* __restrict__ WQ,  // (H, G)
    const float* __restrict__ WK,  // (H, G)
    const float* __restrict__ WV,  // (H, G)
    const float* __restrict__ W0,  // (H,)
    float* __restrict__ out)       // (B, G), pre-zeroed, accumulated atomically
{
    __shared__ float qs[GDIM];
    __shared__ float k2s[GDIM];  // k * log2(e), so exp(q*k) == exp2(q*k2)
    __shared__ float vs[GDIM];

    const int bh = blockIdx.y;
    const int h  = bh % HDIM;
    const int b  = bh / HDIM;
    const int tid = threadIdx.x;

    // ---- Stage 1: build q, k2, v for this (b,h) in LDS ----
    {
        const float* xb = x  + (size_t)b * GDIM;
        const float* wq = WQ + (size_t)h * GDIM;
        const float* wk = WK + (size_t)h * GDIM;
        const float* wv = WV + (size_t)h * GDIM;
        for (int g = tid; g < GDIM; g += 256) {
            float xv = xb[g];
            qs[g]  = wq[g] * xv;
            k2s[g] = wk[g] * xv * LOG2E;
            vs[g]  = wv[g] * xv;
        }
    }
    __syncthreads();

    const int lane = tid & 31;
    const int wave = tid >> 5;
    const int half = lane >> 4;   // 0: lanes 0-15 (j rows 0-7), 1: lanes 16-31 (j rows 8-15)
    const int l15  = lane & 15;   // i within tile

    const int ibase0 = blockIdx.x * 256 + wave * 32;
    const int ibase1 = ibase0 + 16;

    // B operands (q tiles): B[0,n] = q[ibase+n], rest zero.
    v2f Bm0, Bm1;
    Bm0[0] = half ? 0.0f : qs[ibase0 + l15];
    Bm0[1] = 0.0f;
    Bm1[0] = half ? 0.0f : qs[ibase1 + l15];
    Bm1[1] = 0.0f;

    float den0 = 0.0f, num0 = 0.0f;
    float den1 = 0.0f, num1 = 0.0f;

    for (int jt = 0; jt < GDIM / 16; ++jt) {
        const int jbase = jt << 4;

        // A operand (k2 tile): A[m,0] = k2[jbase+m], rest zero.
        v2f Amat;
        Amat[0] = half ? 0.0f : k2s[jbase + l15];
        Amat[1] = 0.0f;

        v8f c0 = {};
        v8f c1 = {};
        c0 = __builtin_amdgcn_wmma_f32_16x16x4_f32(false, Amat, false, Bm0,
                                                   (short)0, c0, false, false);
        c1 = __builtin_amdgcn_wmma_f32_16x16x4_f32(false, Amat, false, Bm1,
                                                   (short)0, c1, false, false);

        // v_j for this lane's j rows: 8 consecutive floats -> two ds_load_b128
        float vj[8];
        const int vbase = jbase + (half << 3);
        *(float4*)&vj[0] = *(const float4*)&vs[vbase];
        *(float4*)&vj[4] = *(const float4*)&vs[vbase + 4];

#pragma unroll
        for (int r = 0; r < 8; ++r) {
            float e0 = __builtin_amdgcn_exp2f(c0[r]);
            den0 += e0;
            num0 = __builtin_fmaf(e0, vj[r], num0);
            float e1 = __builtin_amdgcn_exp2f(c1[r]);
            den1 += e1;
            num1 = __builtin_fmaf(e1, vj[r], num1);
        }
    }

    // Merge the two lane halves (j 0-7 vs 8-15 of every tile)
    den0 += __shfl_xor(den0, 16, 32);
    num0 += __shfl_xor(num0, 16, 32);
    den1 += __shfl_xor(den1, 16, 32);
    num1 += __shfl_xor(num1, 16, 32);

    const float w0h = W0[h];

    {   // tile 0: diagonal correction + normalize + accumulate over h
        const int i = ibase0 + l15;
        float corr = __builtin_amdgcn_exp2f(qs[i] * k2s[i]) * vs[i];
        float a = (num0 - corr) / den0;
        if (lane < 16) atomicAdd(&out[(size_t)b * GDIM + i], w0h * a);
    }
    {   // tile 1
        const int i = ibase1 + l15;
        float corr = __builtin_amdgcn_exp2f(qs[i] * k2s[i]) * vs[i];
        float a = (num1 - corr) / den1;
        if (lane < 16) atomicAdd(&out[(size_t)b * GDIM + i], w0h * a);
    }
}

extern "C" void kernel_launch(void* const* d_in, const int* in_sizes, int n_in,
                              void* d_out, int out_size, void* d_ws, size_t ws_size,
                              hipStream_t stream) {
    (void)in_sizes; (void)n_in; (void)d_ws; (void)ws_size; (void)out_size;

    const float* x  = (const float*)d_in[0];
    const float* WQ = (const float*)d_in[1];
    const float* WK = (const float*)d_in[2];
    const float* WV = (const float*)d_in[3];
    const float* W0 = (const float*)d_in[4];
    float* out = (float*)d_out;

    const int n_out = BDIM * GDIM;
    zero_out_kernel<<<(n_out + 255) / 256, 256, 0, stream>>>(out, n_out);

    dim3 grid(GDIM / 256, BDIM * HDIM);  // (8 i-chunks, 64 (b,h) pairs)
    attn_rank1_softmax_kernel<<<grid, 256, 0, stream>>>(x, WQ, WK, WV, W0, out);
}